// VectorQuantizer_23158463660247
// MI455X (gfx1250) — compile-verified
//
#include <hip/hip_runtime.h>

typedef float v2f __attribute__((ext_vector_type(2)));
typedef float v8f __attribute__((ext_vector_type(8)));

#define B_ 4
#define D_ 256
#define L_ 8192
#define K_ 4096

// ---------------- Kernel 1: e_sq[k] = ||E[k]||^2 ----------------
__global__ void vq_esq(const float* __restrict__ emb, float* __restrict__ esq) {
    const int wave = threadIdx.x >> 5;
    const int lane = threadIdx.x & 31;
    const int k = blockIdx.x * 8 + wave;
    const float* row = emb + (size_t)k * D_;
    float s = 0.f;
#pragma unroll
    for (int i = 0; i < D_ / 32; ++i) {
        float v = row[lane + 32 * i];
        s += v * v;
    }
#pragma unroll
    for (int m = 16; m >= 1; m >>= 1) s += __shfl_xor(s, m, 32);
    if (lane == 0) esq[k] = s;
}

// ---------------- Kernel 2: per-token argmin over K codes via f32 WMMA ----
// score[l][k] = e_sq[k] - 2 * dot(x[:,l], E[k,:])   (||x||^2 dropped: const per token)
// 8 waves/block, 16 tokens/wave; E tiles double-buffered in LDS.
__global__ __launch_bounds__(256, 1) void vq_argmin(const float* __restrict__ x,
                                                    const float* __restrict__ emb,
                                                    const float* __restrict__ esq,
                                                    int* __restrict__ out_idx) {
    // Two E tiles: 16 codes x 256 dims each, row stride 260 floats (bank-conflict free)
    __shared__ float lse[2][16 * 260];

    const int tid  = threadIdx.x;
    const int wave = tid >> 5;
    const int lane = tid & 31;
    const int m    = lane & 15;          // A: row M ; B/C: col N
    const int dOff = (lane >> 4) * 2;    // K sub-offset for upper half-wave

    const int token0 = blockIdx.x * 128 + wave * 16;  // 16 tokens per wave
    const int b  = token0 / L_;
    const int l0 = token0 % L_;
    const float* xb = x + (size_t)b * D_ * L_ + l0 + m;

    // Load A fragments for the whole D=256 reduction: a[s] covers dims 4s..4s+3
    v2f a[64];
#pragma unroll
    for (int s = 0; s < 64; ++s) {
        const int d = 4 * s + dOff;
        a[s].x = xb[(size_t)d * L_];
        a[s].y = xb[(size_t)(d + 1) * L_];
    }

    // Cooperative E-tile copy indexing: c4 constant per thread, row = r0 + 4i
    const int c4 = (tid & 63) * 4;       // float column (16B aligned)
    const int r0 = tid >> 6;             // base code row 0..3
    const float* egbase = emb + (size_t)r0 * D_ + c4;

    // Prologue: stage tile 0 into buffer 0
    {
        float4 p[4];
#pragma unroll
        for (int i = 0; i < 4; ++i) p[i] = *(const float4*)(egbase + (size_t)i * 4 * D_);
#pragma unroll
        for (int i = 0; i < 4; ++i) *(float4*)(&lse[0][(r0 + 4 * i) * 260 + c4]) = p[i];
        __syncthreads();
    }

    float best[8];
    int   bidx[8];
#pragma unroll
    for (int r = 0; r < 8; ++r) { best[r] = 3.4e38f; bidx[r] = 0; }

    for (int t = 0; t < K_ / 16; ++t) {
        const int n0 = t << 4;
        const bool hasNext = (t + 1 < K_ / 16);

        // Prefetch next E tile into registers (overlaps with WMMA below)
        float4 p[4];
        if (hasNext) {
            const float* g = egbase + (size_t)(t + 1) * 16 * D_;
#pragma unroll
            for (int i = 0; i < 4; ++i) p[i] = *(const float4*)(g + (size_t)i * 4 * D_);
        }

        const float e2 = esq[n0 + m];

        // Two independent accumulation chains over the 64 K-steps
        v8f acc0 = {}, acc1 = {};
        const float* bl = &lse[t & 1][m * 260 + dOff];
#pragma unroll
        for (int s = 0; s < 64; s += 2) {
            v2f b0, b1;
            b0.x = bl[4 * s];     b0.y = bl[4 * s + 1];
            b1.x = bl[4 * s + 4]; b1.y = bl[4 * s + 5];
            acc0 = __builtin_amdgcn_wmma_f32_16x16x4_f32(
                false, a[s], false, b0, (short)0, acc0, false, false);
            acc1 = __builtin_amdgcn_wmma_f32_16x16x4_f32(
                false, a[s + 1], false, b1, (short)0, acc1, false, false);
        }

        const int code = n0 + m;
#pragma unroll
        for (int r = 0; r < 8; ++r) {
            const float score = e2 - 2.0f * (acc0[r] + acc1[r]);
            if (score < best[r]) { best[r] = score; bidx[r] = code; }  // first-min wins
        }

        // Commit prefetched tile into the other buffer (read last in iter t-1,
        // fenced by the previous barrier), then one barrier per iteration.
        if (hasNext) {
#pragma unroll
            for (int i = 0; i < 4; ++i)
                *(float4*)(&lse[(t + 1) & 1][(r0 + 4 * i) * 260 + c4]) = p[i];
        }
        __syncthreads();
    }

    // Cross-lane argmin within each 16-lane half (rows r and r+8)
#pragma unroll
    for (int r = 0; r < 8; ++r) {
        float s = best[r];
        int   i = bidx[r];
#pragma unroll
        for (int msk = 8; msk >= 1; msk >>= 1) {
            const float s2 = __shfl_xor(s, msk, 32);
            const int   i2 = __shfl_xor(i, msk, 32);
            if (s2 < s || (s2 == s && i2 < i)) { s = s2; i = i2; }
        }
        if (m == 0) {
            const int tok = token0 + r + (lane >> 4) * 8;
            out_idx[tok] = i;
        }
    }
}

// ---------------- Kernel 3: gather + transpose + int64 indices ------------
__global__ void vq_gather(const float* __restrict__ emb,
                          const int* __restrict__ idx,
                          float* __restrict__ outq,
                          long long* __restrict__ outi) {
    const size_t gid = (size_t)blockIdx.x * 256 + threadIdx.x;  // over B*D*L
    const int l = (int)(gid % L_);
    const int d = (int)((gid / L_) % D_);
    const int b = (int)(gid / ((size_t)D_ * L_));
    const int code = idx[b * L_ + l];
    outq[gid] = emb[(size_t)code * D_ + d];   // coalesced write, L2-cached gather
    if (d == 0) outi[b * L_ + l] = (long long)code;
}

extern "C" void kernel_launch(void* const* d_in, const int* in_sizes, int n_in,
                              void* d_out, int out_size, void* d_ws, size_t ws_size,
                              hipStream_t stream) {
    const float* x   = (const float*)d_in[0];   // [B, D, L] f32
    const float* emb = (const float*)d_in[1];   // [K, D] f32

    float* esq = (float*)d_ws;                               // 4096 f32
    int*   idx = (int*)((char*)d_ws + K_ * sizeof(float));   // 32768 i32

    float*     outq = (float*)d_out;                         // [B, D, L] f32
    long long* outi = (long long*)(outq + (size_t)B_ * D_ * L_);  // [B, L] i64

    vq_esq   <<<K_ / 8,               256, 0, stream>>>(emb, esq);
    vq_argmin<<<(B_ * L_) / 128,      256, 0, stream>>>(x, emb, esq, idx);
    vq_gather<<<(B_ * D_ * L_) / 256, 256, 0, stream>>>(emb, idx, outq, outi);
}